// GCNLayer_9457517986513
// MI455X (gfx1250) — compile-verified
//
#include <hip/hip_runtime.h>
#include <hip/hip_bf16.h>

typedef __attribute__((ext_vector_type(2))) float v2f;
typedef __attribute__((ext_vector_type(8))) float v8f;

#define D 128

// ---------------------------------------------------------------------------
// Native gfx1250 hardware tanh (TRANS op). The trailing v_nop satisfies the
// TRANS-op hazard rule (1 independent op or V_NOP before the result is used),
// since inline asm bypasses the compiler's hazard scheduling.
// ---------------------------------------------------------------------------
__device__ __forceinline__ float hw_tanhf(float x) {
    float r;
    asm volatile("v_tanh_f32 %0, %1\n\tv_nop" : "=v"(r) : "v"(x));
    return r;
}

// ---------------------------------------------------------------------------
// Kernel 1: zero the neighbor-sum accumulator (float4 stores).
// ---------------------------------------------------------------------------
__global__ __launch_bounds__(256) void gcn_zero_kernel(float4* __restrict__ ws,
                                                       int n4) {
    int i = blockIdx.x * blockDim.x + threadIdx.x;
    if (i < n4) {
        ws[i] = make_float4(0.f, 0.f, 0.f, 0.f);
    }
}

// ---------------------------------------------------------------------------
// Kernel 2: edge aggregation. One wave32 per edge; each lane gathers a float4
// (32 lanes x 4 = 128 = D) from node_feats[dst] and scatter-adds into
// nbr_sum[src] with native fp32 global atomics (GLOBAL_ATOMIC_ADD_F32).
// Both the 25.6 MB feature array and the 25.6 MB accumulator fit in the
// 192 MB L2, so this phase runs at L2/atomic-unit speed, not HBM speed.
// ---------------------------------------------------------------------------
__global__ __launch_bounds__(256) void gcn_edge_kernel(
    const float* __restrict__ node_feats,
    const int* __restrict__ edge_index,   // [2, E] flat
    float* __restrict__ nbr_sum,
    int n_edges) {
    const int lane = threadIdx.x & 31;
    const int e = blockIdx.x * (blockDim.x >> 5) + (threadIdx.x >> 5);
    if (e >= n_edges) return;

    const int src = edge_index[e];            // row 0
    const int dst = edge_index[n_edges + e];  // row 1

    const float4 v =
        *(const float4*)(node_feats + (size_t)dst * D + lane * 4);
    float* p = nbr_sum + (size_t)src * D + lane * 4;
    unsafeAtomicAdd(p + 0, v.x);
    unsafeAtomicAdd(p + 1, v.y);
    unsafeAtomicAdd(p + 2, v.z);
    unsafeAtomicAdd(p + 3, v.w);
}

// ---------------------------------------------------------------------------
// Kernel 3: fused dual GEMM + epilogue using V_WMMA_F32_16X16X4_F32.
//   out = node_feats @ self_w + tanh(nbr_sum @ nbr_w + b)
// Block = 256 threads = 8 waves; block b owns node rows [16b, 16b+16);
// wave w owns output columns [16w, 16w+16). K=128 swept in 32 steps of 4.
//
// WMMA fp32 layouts (wave32):
//   A (16x4):  lane&15 = M row; lane>>4 selects K pair {0,1} or {2,3};
//              v2f = A[M][kbase], A[M][kbase+1]
//   B (4x16):  lane&15 = N col; lane>>4 selects K pair; v2f = B[kbase..+1][N]
//   C/D 16x16: VGPR j -> M = j + 8*(lane>>4), N = lane&15
// ---------------------------------------------------------------------------
__global__ __launch_bounds__(256) void gcn_gemm_kernel(
    const float* __restrict__ node_feats,  // [N, 128]
    const float* __restrict__ nbr_sum,     // [N, 128]
    const float* __restrict__ self_w,      // [128, 128]
    const float* __restrict__ nbr_w,       // [128, 128]
    const float* __restrict__ bias,        // [128]
    float* __restrict__ out) {             // [N, 128]
    const int lane  = threadIdx.x & 31;
    const int wave  = threadIdx.x >> 5;    // 0..7 -> column tile
    const int row   = lane & 15;
    const int khalf = lane >> 4;           // 0 or 1 -> K pair within step
    const int m0    = blockIdx.x * 16;
    const int n0    = wave * 16;
    const int col   = lane & 15;

    // A operands: per-lane row pointers, offset to this lane's K pair.
    const float* xrow = node_feats + (size_t)(m0 + row) * D + khalf * 2;
    const float* srow = nbr_sum    + (size_t)(m0 + row) * D + khalf * 2;
    // B operands: per-lane column pointers, offset to this lane's K pair.
    const float* wsp = self_w + (size_t)(khalf * 2) * D + n0 + col;
    const float* wnp = nbr_w  + (size_t)(khalf * 2) * D + n0 + col;

    v8f acc_self = {};
    v8f acc_nbr  = {};

#pragma unroll 4
    for (int kk = 0; kk < D; kk += 4) {
        v2f a_self = *(const v2f*)(xrow + kk);
        v2f a_nbr  = *(const v2f*)(srow + kk);
        v2f b_self;
        b_self.x = wsp[(size_t)kk * D];
        b_self.y = wsp[(size_t)(kk + 1) * D];
        v2f b_nbr;
        b_nbr.x = wnp[(size_t)kk * D];
        b_nbr.y = wnp[(size_t)(kk + 1) * D];

        acc_self = __builtin_amdgcn_wmma_f32_16x16x4_f32(
            false, a_self, false, b_self, (short)0, acc_self, false, false);
        acc_nbr = __builtin_amdgcn_wmma_f32_16x16x4_f32(
            false, a_nbr, false, b_nbr, (short)0, acc_nbr, false, false);
    }

    // Epilogue: out = self + tanh(nbr + b), with hardware v_tanh_f32.
    const float bv = bias[n0 + col];
    float* orow = out + (size_t)(m0 + 8 * khalf) * D + n0 + col;
#pragma unroll
    for (int j = 0; j < 8; ++j) {
        float v = acc_self[j] + hw_tanhf(acc_nbr[j] + bv);
        orow[(size_t)j * D] = v;
    }
}

// ---------------------------------------------------------------------------
extern "C" void kernel_launch(void* const* d_in, const int* in_sizes, int n_in,
                              void* d_out, int out_size, void* d_ws,
                              size_t ws_size, hipStream_t stream) {
    const float* node_feats = (const float*)d_in[0];
    const int*   edge_index = (const int*)d_in[1];
    const float* nbr_w      = (const float*)d_in[2];
    const float* self_w     = (const float*)d_in[3];
    const float* bias       = (const float*)d_in[4];
    float*       out        = (float*)d_out;
    float*       nbr_sum    = (float*)d_ws;   // [N, D] fp32 accumulator

    const int n_nodes = in_sizes[0] / D;      // 50000
    const int n_edges = in_sizes[1] / 2;      // 800000

    // 1) Zero the accumulator (N*D floats = N*32 float4s).
    {
        const int n4 = n_nodes * (D / 4);
        gcn_zero_kernel<<<(n4 + 255) / 256, 256, 0, stream>>>(
            (float4*)nbr_sum, n4);
    }

    // 2) Edge scatter-add: one wave per edge, 8 waves per block.
    {
        const int blocks = (n_edges + 7) / 8;
        gcn_edge_kernel<<<blocks, 256, 0, stream>>>(node_feats, edge_index,
                                                    nbr_sum, n_edges);
    }

    // 3) Fused dual WMMA GEMM + tanh epilogue. 50000/16 = 3125 row tiles.
    {
        const int blocks = n_nodes / 16;
        gcn_gemm_kernel<<<blocks, 256, 0, stream>>>(node_feats, nbr_sum,
                                                    self_w, nbr_w, bias, out);
    }
}